// SelfAttentionBlock_6983616823332
// MI455X (gfx1250) — compile-verified
//
#include <hip/hip_runtime.h>
#include <math.h>
#include <stdint.h>

#define EDIM 256
#define HEADS 8
#define NTOK 1500
#define BATCH 4
#define DHEAD 32
#define HID 512
#define NT 6000            // BATCH*NTOK tokens
#define EPSLN 1e-6f
#define KC 64              // K-chunk staged through LDS

typedef __attribute__((ext_vector_type(2))) float v2f;
typedef __attribute__((ext_vector_type(8))) float v8f;
typedef __attribute__((ext_vector_type(4))) int v4i;
typedef __attribute__((address_space(1))) v4i* gv4i_t;  // global int4*
typedef __attribute__((address_space(3))) v4i* lv4i_t;  // LDS int4*

// D = A(16x4) * B(4x16) + C, full fp32 (CDNA5 V_WMMA_F32_16X16X4_F32)
#define WMMA4(a, b, c) \
  __builtin_amdgcn_wmma_f32_16x16x4_f32(false, (a), false, (b), (short)0, (c), false, false)

// ---- async global -> LDS copy (16B per lane), tracked with ASYNCcnt --------
__device__ __forceinline__ void async_copy_b128(const float* gsrc, float* lds_dst) {
#if __has_builtin(__builtin_amdgcn_global_load_async_to_lds_b128)
  __builtin_amdgcn_global_load_async_to_lds_b128((gv4i_t)gsrc, (lv4i_t)lds_dst, 0, 0);
#else
  const unsigned lds_off = (unsigned)(uintptr_t)lds_dst;  // low 32 bits = LDS offset
  asm volatile("global_load_async_to_lds_b128 %0, %1, off"
               :: "v"(lds_off), "v"(gsrc) : "memory");
#endif
}

template <int N>
__device__ __forceinline__ void wait_asynccnt() {
#if __has_builtin(__builtin_amdgcn_s_wait_asynccnt)
  __builtin_amdgcn_s_wait_asynccnt(N);
#else
  if constexpr (N == 0) {
    asm volatile("s_wait_asynccnt 0x0" ::: "memory");
  } else {
    asm volatile("s_wait_asynccnt 0x1" ::: "memory");
  }
#endif
}

// ---------------------------------------------------------------------------
// LayerNorm over E=256, one wave per token; optionally beta[b,h,n].
// ---------------------------------------------------------------------------
template <bool DO_BETA>
__global__ __launch_bounds__(256) void ln_kernel(
    const float* __restrict__ feat, const float* __restrict__ w,
    const float* __restrict__ bias, const float* __restrict__ beta_w,
    const float* __restrict__ beta_b, float* __restrict__ x_out,
    float* __restrict__ beta_out) {
  const int wave = threadIdx.x >> 5;
  const int lane = threadIdx.x & 31;
  const int tok = blockIdx.x * 8 + wave;          // grid = 750 -> 6000 tokens
  const float* row = feat + (size_t)tok * EDIM;

  float v[8];
  float s = 0.f;
#pragma unroll
  for (int i = 0; i < 8; ++i) { v[i] = row[lane + 32 * i]; s += v[i]; }
#pragma unroll
  for (int m = 16; m >= 1; m >>= 1) s += __shfl_xor(s, m, 32);
  const float mu = s * (1.f / 256.f);
  float var = 0.f;
#pragma unroll
  for (int i = 0; i < 8; ++i) { float d = v[i] - mu; var += d * d; }
#pragma unroll
  for (int m = 16; m >= 1; m >>= 1) var += __shfl_xor(var, m, 32);
  const float rstd = rsqrtf(var * (1.f / 256.f) + EPSLN);

  float xo[8];
#pragma unroll
  for (int i = 0; i < 8; ++i) {
    const int e = lane + 32 * i;
    xo[i] = (v[i] - mu) * rstd * w[e] + bias[e];
    x_out[(size_t)tok * EDIM + e] = xo[i];
  }

  if (DO_BETA) {
    const int b = tok / NTOK, n = tok % NTOK;
#pragma unroll
    for (int h = 0; h < HEADS; ++h) {
      float p = 0.f;
#pragma unroll
      for (int i = 0; i < 8; ++i) p += xo[i] * beta_w[h * EDIM + lane + 32 * i];
#pragma unroll
      for (int m = 16; m >= 1; m >>= 1) p += __shfl_xor(p, m, 32);
      if (lane == 0) beta_out[(size_t)(b * HEADS + h) * NTOK + n] = p + beta_b[h];
    }
  }
}

// ---------------------------------------------------------------------------
// C[M,Nc] = A[M,K] @ W[Nc,K]^T + bias  (+ epilogue)
// EPI: 0 = bias, 1 = bias + residual, 2 = bias + exact GELU
// Block = 8 waves: one 16-row m-tile, 8 consecutive 16-col n-tiles.
// The shared 16xKC A-chunk is staged via async global->LDS copies
// (double-buffered, ASYNCcnt-tracked); W fragments come from L2.
// grid = (Nc/128, M/16), exact.  M,Nc mult of 16; K mult of KC.
// ---------------------------------------------------------------------------
template <int EPI>
__global__ __launch_bounds__(256) void wmma_gemm_kernel(
    const float* __restrict__ A, const float* __restrict__ W,
    const float* __restrict__ bias, const float* __restrict__ resid,
    float* __restrict__ C, int M, int Nc, int K) {
  __shared__ float As[2][16 * 68];  // 68-float row stride: conflict-free banks
  const int tid = threadIdx.x;
  const int lane = tid & 31;
  const int wave = tid >> 5;
  const int hi = lane >> 4;
  const int l15 = lane & 15;
  const int n0 = (blockIdx.x * 8 + wave) * 16;
  const int m0 = blockIdx.y * 16;

  // this thread's float4 of each A chunk (perfectly coalesced)
  const int arow = tid >> 4;        // 0..15
  const int acol = (tid & 15) * 4;  // 0..60
  const float* agsrc = A + (size_t)(m0 + arow) * K + acol;
  float* adst0 = &As[0][arow * 68 + acol];
  float* adst1 = &As[1][arow * 68 + acol];

  const float* brow = W + (size_t)(n0 + l15) * K + 2 * hi;

  async_copy_b128(agsrc, adst0);  // prologue: chunk 0 -> buffer 0

  v8f acc = {};
  const int nchunks = K / KC;
  for (int c = 0; c < nchunks; ++c) {
    if (c + 1 < nchunks) {
      async_copy_b128(agsrc + (size_t)(c + 1) * KC, (c & 1) ? adst0 : adst1);
      wait_asynccnt<1>();  // chunk c landed; chunk c+1 still in flight
    } else {
      wait_asynccnt<0>();
    }
    __syncthreads();
    const float* ap = &As[c & 1][l15 * 68 + 2 * hi];
    const float* bp = brow + c * KC;
#pragma unroll
    for (int k = 0; k < KC; k += 4) {
      v2f a = *(const v2f*)(ap + k);   // LDS: A[m0+l15][kc+k+2*hi+{0,1}]
      v2f b = *(const v2f*)(bp + k);   // B[k'][n] = W[n0+l15][kc+k+2*hi+{0,1}]
      acc = WMMA4(a, b, acc);
    }
    __syncthreads();  // finish reads before this buffer is overwritten
  }

  const float bn = bias[n0 + l15];
#pragma unroll
  for (int r = 0; r < 8; ++r) {
    const int m = m0 + r + 8 * hi;
    float v = acc[r] + bn;
    if (EPI == 1) v += resid[(size_t)m * Nc + n0 + l15];
    if (EPI == 2) v = 0.5f * v * (1.f + erff(v * 0.70710678118654752f));
    C[(size_t)m * Nc + n0 + l15] = v;
  }
}

// ---------------------------------------------------------------------------
// Flash attention with distance-bias mask, one wave per (batch, head, q-tile).
// qkv: (B, N, 768) with cols [0,256)=Q, [256,512)=K, [512,768)=V (head-major).
// S^T = K_tile @ Q^T (softmax rows live per-lane), O^T = V^T @ P^T with
// P^T remapped C-layout -> B-fragments via shfl_xor(16).
// block = 256 threads = 8 waves = 8 heads; grid = (ceil(N/16), B).
// ---------------------------------------------------------------------------
__global__ __launch_bounds__(256) void attn_kernel(
    const float* __restrict__ qkv, const float* __restrict__ points,
    const float* __restrict__ beta, float* __restrict__ attn_out) {
  const int h = threadIdx.x >> 5;
  const int lane = threadIdx.x & 31;
  const int hi = lane >> 4;
  const int l15 = lane & 15;
  const int b = blockIdx.y;
  const int q0 = blockIdx.x * 16;
  const int q = q0 + l15;
  const int qc = min(q, NTOK - 1);

  const float* Qrow = qkv + (size_t)(b * NTOK + qc) * 768 + h * DHEAD;
  v2f qb[8];
#pragma unroll
  for (int s = 0; s < 8; ++s) qb[s] = *(const v2f*)(Qrow + 4 * s + 2 * hi);

  const float cqx = points[(size_t)(b * NTOK + qc) * 3 + 0];
  const float cqy = points[(size_t)(b * NTOK + qc) * 3 + 1];
  const float betav = beta[(size_t)(b * HEADS + h) * NTOK + qc];
  const float scal = 0.17677669529663687f;  // 1/sqrt(32)

  float m_run = -1e30f, l_run = 0.f;
  v8f o0 = {}, o1 = {};
  const float* Kbase = qkv + (size_t)b * NTOK * 768 + 256 + h * DHEAD;
  const float* Vbase = qkv + (size_t)b * NTOK * 768 + 512 + h * DHEAD;

  for (int k0 = 0; k0 < NTOK; k0 += 16) {
    // ---- S^T tile: rows = keys, cols = queries ----
    const int kac = min(k0 + l15, NTOK - 1);
    const float* Krow = Kbase + (size_t)kac * 768;
    v8f st = {};
#pragma unroll
    for (int s = 0; s < 8; ++s) {
      v2f a = *(const v2f*)(Krow + 4 * s + 2 * hi);  // K[key=l15][d]
      st = WMMA4(a, qb[s], st);
    }

    // ---- scale + distance-bias mask ----
    const float ckx = points[(size_t)(b * NTOK + kac) * 3 + 0];
    const float cky = points[(size_t)(b * NTOK + kac) * 3 + 1];
#pragma unroll
    for (int r = 0; r < 8; ++r) {
      const int key = k0 + r + 8 * hi;
      const float kx = __shfl(ckx, r + 8 * hi, 32);
      const float ky = __shfl(cky, r + 8 * hi, 32);
      const float dx = cqx - kx, dy = cqy - ky;
      const float dist = -sqrtf(dx * dx + dy * dy);
      float v = st[r] * scal + dist * betav;
      if (key >= NTOK) v = -1e30f;  // pad keys
      st[r] = v;
    }

    // ---- online softmax (row = fixed q = fixed lane15, halves share q) ----
    float mx = st[0];
#pragma unroll
    for (int r = 1; r < 8; ++r) mx = fmaxf(mx, st[r]);
    mx = fmaxf(mx, __shfl_xor(mx, 16, 32));
    const float m_new = fmaxf(m_run, mx);
    const float rescale = expf(m_run - m_new);
    float lsum = 0.f;
#pragma unroll
    for (int r = 0; r < 8; ++r) {
      const float p = expf(st[r] - m_new);
      st[r] = p;
      lsum += p;
    }
    lsum += __shfl_xor(lsum, 16, 32);
    l_run = l_run * rescale + lsum;
    m_run = m_new;
#pragma unroll
    for (int r = 0; r < 8; ++r) { o0[r] *= rescale; o1[r] *= rescale; }

    // ---- O^T += V^T @ P^T ----
#pragma unroll
    for (int s = 0; s < 4; ++s) {
      const int base = (s & 1) * 4;
      v2f pb;
#pragma unroll
      for (int j = 0; j < 2; ++j) {
        const float t_lo = st[base + j];
        const float t_hi = st[base + j + 2];
        const float x_lo = __shfl_xor(t_lo, 16, 32);
        const float x_hi = __shfl_xor(t_hi, 16, 32);
        pb[j] = (s < 2) ? (hi ? x_hi : t_lo) : (hi ? t_hi : x_lo);
      }
      const int keyx = k0 + 4 * s + 2 * hi;
      const int k1c = min(keyx, NTOK - 1);
      const int k2c = min(keyx + 1, NTOK - 1);
      const float* vr1 = Vbase + (size_t)k1c * 768;
      const float* vr2 = Vbase + (size_t)k2c * 768;
      v2f a0 = {vr1[l15], vr2[l15]};            // V^T[d=l15][key]
      v2f a1 = {vr1[16 + l15], vr2[16 + l15]};  // V^T[d=16+l15][key]
      o0 = WMMA4(a0, pb, o0);
      o1 = WMMA4(a1, pb, o1);
    }
  }

  if (q < NTOK) {
    const float inv = 1.f / l_run;
    float* orow = attn_out + (size_t)(b * NTOK + q) * EDIM + h * DHEAD + 8 * hi;
#pragma unroll
    for (int r = 0; r < 8; ++r) {
      orow[r] = o0[r] * inv;        // d = r + 8*hi
      orow[16 + r] = o1[r] * inv;   // d = 16 + r + 8*hi
    }
  }
}

// ---------------------------------------------------------------------------
extern "C" void kernel_launch(void* const* d_in, const int* in_sizes, int n_in,
                              void* d_out, int out_size, void* d_ws,
                              size_t ws_size, hipStream_t stream) {
  const float* feat = (const float*)d_in[0];
  const float* points = (const float*)d_in[1];
  const float* ln1_w = (const float*)d_in[2];
  const float* ln1_b = (const float*)d_in[3];
  const float* beta_w = (const float*)d_in[4];
  const float* beta_b = (const float*)d_in[5];
  const float* in_w = (const float*)d_in[6];
  const float* in_b = (const float*)d_in[7];
  const float* out_w = (const float*)d_in[8];
  const float* out_b = (const float*)d_in[9];
  const float* ln2_w = (const float*)d_in[10];
  const float* ln2_b = (const float*)d_in[11];
  const float* fc1_w = (const float*)d_in[12];
  const float* fc1_b = (const float*)d_in[13];
  const float* fc2_w = (const float*)d_in[14];
  const float* fc2_b = (const float*)d_in[15];

  float* ws = (float*)d_ws;
  float* x_ln = ws;                 // 6000*256      = 1,536,000 f
  float* betab = ws + 1536000;      // 4*8*1500      =    48,000 f
  float* qkv = ws + 1584000;        // 6000*768      = 4,608,000 f
  float* attn = ws + 6192000;       // 6000*256      = 1,536,000 f
  float* feat2 = ws + 7728000;      // 6000*256      = 1,536,000 f
  float* y2 = attn;                 // reuse after out-proj consumed attn
  float* hfc1 = qkv;                // reuse after attention consumed qkv
  float* outp = (float*)d_out;

  // 1) LN1 + beta projection
  ln_kernel<true><<<dim3(750), dim3(256), 0, stream>>>(
      feat, ln1_w, ln1_b, beta_w, beta_b, x_ln, betab);
  // 2) QKV projection: (6000x256)@(768x256)^T
  wmma_gemm_kernel<0><<<dim3(6, 375), dim3(256), 0, stream>>>(
      x_ln, in_w, in_b, nullptr, qkv, NT, 768, 256);
  // 3) flash attention with distance-bias
  attn_kernel<<<dim3(94, BATCH), dim3(256), 0, stream>>>(qkv, points, betab, attn);
  // 4) output projection + residual
  wmma_gemm_kernel<1><<<dim3(2, 375), dim3(256), 0, stream>>>(
      attn, out_w, out_b, feat, feat2, NT, 256, 256);
  // 5) LN2
  ln_kernel<false><<<dim3(750), dim3(256), 0, stream>>>(
      feat2, ln2_w, ln2_b, nullptr, nullptr, y2, nullptr);
  // 6) FC1 + exact GELU: (6000x256)@(512x256)^T
  wmma_gemm_kernel<2><<<dim3(4, 375), dim3(256), 0, stream>>>(
      y2, fc1_w, fc1_b, nullptr, hfc1, NT, 512, 256);
  // 7) FC2 -> out: (6000x512)@(256x512)^T
  wmma_gemm_kernel<0><<<dim3(2, 375), dim3(256), 0, stream>>>(
      hfc1, fc2_w, fc2_b, nullptr, outp, NT, 256, 512);
}